// New_GAT_LSTM_89008902243184
// MI455X (gfx1250) — compile-verified
//
#include <hip/hip_runtime.h>
#include <math.h>

// ---------------------------------------------------------------------------
// GAT(2 layer) + LSTM + node-softmax for MI455X (gfx1250, wave32).
// Dense GEMMs: V_WMMA_F32_16X16X4_F32 fed from an LDS A-tile staged with
// GLOBAL_LOAD_ASYNC_TO_LDS_B128 (ASYNCcnt) shared by all waves of the block.
// Sparse edge softmax/aggregation: f32 global atomics (L2-resident) with
// GLOBAL_PREFETCH on the gathered feature rows.
// ---------------------------------------------------------------------------

typedef __attribute__((ext_vector_type(2))) float v2f;
typedef __attribute__((ext_vector_type(8))) float v8f;

#define NN 20000            // nodes
#define NE 320000           // edges (without self loops)
#define ET (NE + NN)        // edges + self loops
#define FD 32               // feature dim
#define NH 8                // heads (layer 1)
#define HF 256              // NH*FD
#define CD 32               // LSTM hidden
#define TT 8                // timesteps
#define SLOPE 0.2f

__device__ __forceinline__ float sigf(float x) { return 1.f / (1.f + __expf(-x)); }

__device__ __forceinline__ void atomic_max_f32(float* addr, float val) {
  int* ia = (int*)addr;
  int old = *ia;
  while (__int_as_float(old) < val) {
    int assumed = old;
    old = atomicCAS(ia, assumed, __float_as_int(val));
    if (old == assumed) break;
  }
}

__global__ void fill_k(float* p, int n, float v) {
  int i = blockIdx.x * blockDim.x + threadIdx.x;
  if (i < n) p[i] = v;
}

// C[M,ncols] = A[M,K] @ B[K,ncols], row-major. One block = 16 output rows.
// The 16xK A-tile is staged once into LDS via async B128 loads and shared by
// all waves; each wave owns column tiles ct = wave, wave+nwaves, ...
// K must be a multiple of 4; rows padded by 4 floats to spread LDS banks.
__global__ void wmma_gemm_k(const float* __restrict__ A, const float* __restrict__ B,
                            float* __restrict__ C, int K, int ncols) {
  extern __shared__ float Atile[];          // 16 * (K+4) floats
  const int KP = K + 4;
  const int r0 = blockIdx.x << 4;

  // ---- async stage A[r0:r0+16, 0:K] -> LDS (B128 per lane) ----
  {
    const int nvec = (K >> 2);              // float4 per row
    for (int i = threadIdx.x; i < 16 * nvec; i += blockDim.x) {
      int row = i / nvec;
      int cv = i - row * nvec;
      unsigned long long gsrc =
          (unsigned long long)(uintptr_t)(A + (size_t)(r0 + row) * K + (cv << 2));
      // flat shared pointer: low 32 bits are the LDS byte offset (aperture rule)
      unsigned int ldsa = (unsigned int)(uintptr_t)(&Atile[row * KP + (cv << 2)]);
      asm volatile("global_load_async_to_lds_b128 %0, %1, off"
                   :: "v"(ldsa), "v"(gsrc) : "memory");
    }
  }
  asm volatile("s_wait_asynccnt 0x0" ::: "memory");
  __syncthreads();

  const int wave = threadIdx.x >> 5;
  const int nwaves = blockDim.x >> 5;
  const int lane = threadIdx.x & 31;
  const int l = lane & 15;
  const int kh = (lane >> 4) << 1;          // 0 for lanes 0-15, 2 for 16-31
  const int ntiles = ncols >> 4;

  for (int ct = wave; ct < ntiles; ct += nwaves) {
    int c0 = ct << 4;
    v8f acc = {};
    for (int k = 0; k < K; k += 4) {
      v2f a, b;
      a.x = Atile[l * KP + k + kh];
      a.y = Atile[l * KP + k + kh + 1];
      b.x = B[(k + kh) * ncols + c0 + l];
      b.y = B[(k + kh + 1) * ncols + c0 + l];
      acc = __builtin_amdgcn_wmma_f32_16x16x4_f32(false, a, false, b, (short)0, acc,
                                                  false, false);
    }
    int rb = r0 + ((lane >> 4) << 3);
#pragma unroll
    for (int r = 0; r < 8; ++r)
      C[(rb + r) * ncols + c0 + l] = acc[r];
  }
}

// Per-(node,head) attention logits: alsrc/aldst[n,h] = <h[n,h,:], a_src/dst[h,:]>
__global__ void al_k(const float* __restrict__ h, const float* __restrict__ asrc,
                     const float* __restrict__ adst, float* __restrict__ alsrc,
                     float* __restrict__ aldst, int heads, int fdim) {
  int i = blockIdx.x * blockDim.x + threadIdx.x;
  if (i >= NN * heads) return;
  int n = i / heads, hh = i % heads;
  const float* hp = h + (n * heads + hh) * fdim;
  const float* as = asrc + hh * fdim;
  const float* ad = adst + hh * fdim;
  float s1 = 0.f, s2 = 0.f;
  for (int f = 0; f < fdim; ++f) { float v = hp[f]; s1 += v * as[f]; s2 += v * ad[f]; }
  alsrc[i] = s1;
  aldst[i] = s2;
}

__device__ __forceinline__ void edge_sd(const int* ei, int e, int& s, int& d) {
  if (e < NE) { s = ei[e]; d = ei[NE + e]; } else { s = e - NE; d = s; }
}

// segment-max of leaky_relu(alsrc[src]+aldst[dst]) per (dst,head)
__global__ void edge_max_k(const int* __restrict__ ei, const float* __restrict__ alsrc,
                           const float* __restrict__ aldst, float* __restrict__ m,
                           int heads) {
  int i = blockIdx.x * blockDim.x + threadIdx.x;
  if (i >= ET * heads) return;
  int e = i / heads, h = i % heads;
  int s, d; edge_sd(ei, e, s, d);
  float a = alsrc[s * heads + h] + aldst[d * heads + h];
  a = (a > 0.f) ? a : SLOPE * a;
  atomic_max_f32(&m[d * heads + h], a);
}

// denom[dst,h] += exp(alpha-m);  acc[dst,h,:] += exp(alpha-m) * h[src,h,:]
__global__ void edge_acc_k(const int* __restrict__ ei, const float* __restrict__ alsrc,
                           const float* __restrict__ aldst, const float* __restrict__ m,
                           float* __restrict__ den, const float* __restrict__ hfeat,
                           float* __restrict__ acc, int heads, int fdim) {
  int i = blockIdx.x * blockDim.x + threadIdx.x;
  if (i >= ET * heads) return;
  int e = i / heads, h = i % heads;
  int s, d; edge_sd(ei, e, s, d);
  const float* hs = hfeat + (s * heads + h) * fdim;
  __builtin_prefetch(hs, 0, 0);           // global_prefetch_b8: feature row in flight
  float a = alsrc[s * heads + h] + aldst[d * heads + h];
  a = (a > 0.f) ? a : SLOPE * a;
  float ex = __expf(a - m[d * heads + h]);
  atomicAdd(&den[d * heads + h], ex);
  float* ac = acc + (d * heads + h) * fdim;
  for (int f = 0; f < fdim; ++f) atomicAdd(&ac[f], ex * hs[f]);
}

// acc[n,j] = (acc[n,j]/(den[n,h]+1e-16) + bias[j]) [, relu]   (in place)
__global__ void node_div_k(float* __restrict__ acc, const float* __restrict__ den,
                           const float* __restrict__ bias, int heads, int fdim,
                           int relu) {
  int i = blockIdx.x * blockDim.x + threadIdx.x;
  int hf = heads * fdim;
  if (i >= NN * hf) return;
  int j = i % hf;
  int n = i / hf;
  int h = j / fdim;
  float v = acc[i] / (den[n * heads + h] + 1e-16f) + bias[j];
  if (relu) v = fmaxf(v, 0.f);
  acc[i] = v;
}

// One LSTM step. Block = 16 rows; 8 waves each own one 16-col tile of the
// [16,128] gate matrix (two K=32 WMMA chains: x@Wih^T and h@Whh^T), staged
// through LDS, then elementwise gates. out_t may alias xt (reads precede
// the barrier; each block touches only its own rows).
__global__ void lstm_step_k(const float* __restrict__ xt, const float* __restrict__ w_ih,
                            const float* __restrict__ w_hh, const float* __restrict__ b_ih,
                            const float* __restrict__ b_hh, float* __restrict__ hstate,
                            float* __restrict__ cstate, float* __restrict__ out_t) {
  __shared__ float g[16 * 128];
  int r0 = blockIdx.x << 4;
  int wave = threadIdx.x >> 5;
  int lane = threadIdx.x & 31;
  int l = lane & 15;
  int kh = (lane >> 4) << 1;
  int c0 = wave << 4;
  v8f acc = {};
#pragma unroll
  for (int k = 0; k < FD; k += 4) {   // xt @ w_ih^T : B[k,col] = w_ih[col*32+k]
    v2f a, b;
    a.x = xt[(r0 + l) * FD + k + kh];
    a.y = xt[(r0 + l) * FD + k + kh + 1];
    b.x = w_ih[(c0 + l) * FD + k + kh];
    b.y = w_ih[(c0 + l) * FD + k + kh + 1];
    acc = __builtin_amdgcn_wmma_f32_16x16x4_f32(false, a, false, b, (short)0, acc,
                                                false, false);
  }
#pragma unroll
  for (int k = 0; k < CD; k += 4) {   // hprev @ w_hh^T
    v2f a, b;
    a.x = hstate[(r0 + l) * CD + k + kh];
    a.y = hstate[(r0 + l) * CD + k + kh + 1];
    b.x = w_hh[(c0 + l) * CD + k + kh];
    b.y = w_hh[(c0 + l) * CD + k + kh + 1];
    acc = __builtin_amdgcn_wmma_f32_16x16x4_f32(false, a, false, b, (short)0, acc,
                                                false, false);
  }
  int col = c0 + l;
  float bias = b_ih[col] + b_hh[col];
  int rb = (lane >> 4) << 3;
#pragma unroll
  for (int r = 0; r < 8; ++r) g[(rb + r) * 128 + col] = acc[r] + bias;
  __syncthreads();
  for (int item = threadIdx.x; item < 16 * CD; item += blockDim.x) {
    int rl = item >> 5;
    int c = item & 31;
    const float* gr = &g[rl * 128];
    float gi = gr[c], gf = gr[32 + c], gg = gr[64 + c], go = gr[96 + c];
    int n = r0 + rl;
    float cp = cstate[n * CD + c];
    float cn = sigf(gf) * cp + sigf(gi) * tanhf(gg);
    float hn = sigf(go) * tanhf(cn);
    cstate[n * CD + c] = cn;
    hstate[n * CD + c] = hn;
    out_t[n * CD + c] = hn;
  }
}

// softmax over the node dimension: one block per (t, c) column of length NN.
__global__ void softmax_nodes_k(float* __restrict__ out) {
  __shared__ float red[256];
  int b = blockIdx.x;
  int t = b >> 5, c = b & 31;
  float* base = out + (size_t)t * NN * CD + c;
  int tid = threadIdx.x;
  float m = -1e30f;
  for (int n = tid; n < NN; n += 256) m = fmaxf(m, base[n * CD]);
  red[tid] = m; __syncthreads();
  for (int s = 128; s > 0; s >>= 1) { if (tid < s) red[tid] = fmaxf(red[tid], red[tid + s]); __syncthreads(); }
  m = red[0]; __syncthreads();
  float sum = 0.f;
  for (int n = tid; n < NN; n += 256) sum += __expf(base[n * CD] - m);
  red[tid] = sum; __syncthreads();
  for (int s = 128; s > 0; s >>= 1) { if (tid < s) red[tid] += red[tid + s]; __syncthreads(); }
  float inv = 1.f / red[0];
  for (int n = tid; n < NN; n += 256) base[n * CD] = __expf(base[n * CD] - m) * inv;
}

extern "C" void kernel_launch(void* const* d_in, const int* in_sizes, int n_in,
                              void* d_out, int out_size, void* d_ws, size_t ws_size,
                              hipStream_t stream) {
  (void)in_sizes; (void)n_in; (void)out_size; (void)ws_size;
  const float* x    = (const float*)d_in[0];
  const float* W1   = (const float*)d_in[1];
  const float* as1  = (const float*)d_in[2];
  const float* ad1  = (const float*)d_in[3];
  const float* b1   = (const float*)d_in[4];
  const float* W2   = (const float*)d_in[5];
  const float* as2  = (const float*)d_in[6];
  const float* ad2  = (const float*)d_in[7];
  const float* b2   = (const float*)d_in[8];
  const float* w_ih = (const float*)d_in[9];
  const float* w_hh = (const float*)d_in[10];
  const float* b_ih = (const float*)d_in[11];
  const float* b_hh = (const float*)d_in[12];
  const int*   ei   = (const int*)d_in[13];
  float* out = (float*)d_out;

  float* ws = (float*)d_ws;
  size_t off = 0;
  float* h1     = ws + off; off += (size_t)NN * HF;
  float* alsrc1 = ws + off; off += (size_t)NN * NH;
  float* aldst1 = ws + off; off += (size_t)NN * NH;
  float* m1     = ws + off; off += (size_t)NN * NH;
  float* den1   = ws + off; off += (size_t)NN * NH;
  float* acc1   = ws + off; off += (size_t)NN * HF;
  float* h2     = ws + off; off += (size_t)NN * FD;
  float* alsrc2 = ws + off; off += (size_t)NN;
  float* aldst2 = ws + off; off += (size_t)NN;
  float* m2     = ws + off; off += (size_t)NN;
  float* den2   = ws + off; off += (size_t)NN;
  float* hstate = ws + off; off += (size_t)NN * CD;
  float* cstate = ws + off; off += (size_t)NN * CD;

  const int RT = NN / 16;  // 1250 row tiles
  const size_t smem1 = (size_t)16 * (FD + 4) * sizeof(float);   // GEMM1 A-tile
  const size_t smem2 = (size_t)16 * (HF + 4) * sizeof(float);   // GEMM2 A-tile
  auto fill = [&](float* p, int n, float v) {
    fill_k<<<(n + 255) / 256, 256, 0, stream>>>(p, n, v);
  };

  for (int t = 0; t < TT; ++t) {
    const float* xt = x + (size_t)t * NN * FD;
    float* hs_t = out + (size_t)t * NN * CD;

    // ---- GAT layer 1 (heads=8) ----
    fill(m1, NN * NH, -1e30f);
    fill(den1, NN * NH, 0.f);
    fill(acc1, NN * HF, 0.f);
    wmma_gemm_k<<<RT, 256, smem1, stream>>>(xt, W1, h1, FD, HF);
    al_k<<<(NN * NH + 255) / 256, 256, 0, stream>>>(h1, as1, ad1, alsrc1, aldst1, NH, FD);
    edge_max_k<<<(ET * NH + 255) / 256, 256, 0, stream>>>(ei, alsrc1, aldst1, m1, NH);
    edge_acc_k<<<(ET * NH + 255) / 256, 256, 0, stream>>>(ei, alsrc1, aldst1, m1, den1,
                                                          h1, acc1, NH, FD);
    node_div_k<<<(NN * HF + 255) / 256, 256, 0, stream>>>(acc1, den1, b1, NH, FD, 1);

    // ---- GAT layer 2 (heads=1), accumulate straight into out[t] ----
    fill(m2, NN, -1e30f);
    fill(den2, NN, 0.f);
    fill(hs_t, NN * CD, 0.f);
    wmma_gemm_k<<<RT, 64, smem2, stream>>>(acc1, W2, h2, HF, FD);
    al_k<<<(NN + 255) / 256, 256, 0, stream>>>(h2, as2, ad2, alsrc2, aldst2, 1, FD);
    edge_max_k<<<(ET + 255) / 256, 256, 0, stream>>>(ei, alsrc2, aldst2, m2, 1);
    edge_acc_k<<<(ET + 255) / 256, 256, 0, stream>>>(ei, alsrc2, aldst2, m2, den2,
                                                     h2, hs_t, 1, FD);
    node_div_k<<<(NN * FD + 255) / 256, 256, 0, stream>>>(hs_t, den2, b2, 1, FD, 0);
  }

  // ---- LSTM over T, in place on d_out ----
  fill(hstate, NN * CD, 0.f);
  fill(cstate, NN * CD, 0.f);
  for (int t = 0; t < TT; ++t) {
    float* hs_t = out + (size_t)t * NN * CD;
    lstm_step_k<<<RT, 256, 0, stream>>>(hs_t, w_ih, w_hh, b_ih, b_hh, hstate, cstate, hs_t);
  }

  // ---- softmax over nodes, per (t, channel) ----
  softmax_nodes_k<<<TT * CD, 256, 0, stream>>>(out);
}